// GAT_84894323573128
// MI455X (gfx1250) — compile-verified
//
#include <hip/hip_runtime.h>

typedef __attribute__((ext_vector_type(2))) float v2f;
typedef __attribute__((ext_vector_type(8))) float v8f;

// ---------------------------------------------------------------------------
// GEMM: xs = X @ Ws, xd = X @ Wd  (X:[n,64], Ws/Wd:[64,64] row-major)
// V_WMMA_F32_16X16X4_F32, K=64 in 16 steps. Weights staged in LDS, swizzled
// into per-fragment float2 pairs so each B frag is a single ds_load_b64:
//   ldsW[t][c][h] = { W[(4t+2h)][c], W[(4t+2h+1)][c] },  t=kstep, c=col, h=lane half
// A frag (16x4 f32): lane<16 -> K={k,k+1}, lane>=16 -> K={k+2,k+3}, M=lane&15
// C/D: vgpr j -> M = j + 8*(lane>=16), N = lane&15 (per 16-col tile)
// k-loop unroll capped at 2 to keep B frags out of spill territory
// (full unroll + grid-stride made LICM hoist 256 VGPRs of frags -> scratch).
// ---------------------------------------------------------------------------
#define GEMM_BLOCKS 196

__global__ void gat_gemm_wmma(const float* __restrict__ X,
                              const float* __restrict__ Ws,
                              const float* __restrict__ Wd,
                              float* __restrict__ xs,
                              float* __restrict__ xd,
                              int n) {
    __shared__ v2f ldsS[16 * 64 * 2];   // 16 KB
    __shared__ v2f ldsD[16 * 64 * 2];   // 16 KB

    // ---- stage + swizzle both weight matrices into LDS (256 threads) ----
    {
        const int tid = threadIdx.x;
#pragma unroll
        for (int i = 0; i < 8; ++i) {
            const int id = tid + i * 256;          // 0..2047
            const int h = id & 1;
            const int c = (id >> 1) & 63;
            const int t = id >> 7;
            const int r = 4 * t + 2 * h;
            v2f vs, vd;
            vs.x = Ws[(size_t)r * 64 + c];
            vs.y = Ws[(size_t)(r + 1) * 64 + c];
            vd.x = Wd[(size_t)r * 64 + c];
            vd.y = Wd[(size_t)(r + 1) * 64 + c];
            ldsS[id] = vs;
            ldsD[id] = vd;
        }
    }
    __syncthreads();

    const int lane = threadIdx.x & 31;
    const int half = lane >> 4;          // 0 or 1
    const int lan  = lane & 15;
    const int tiles  = (n + 15) >> 4;
    const int gwaves = GEMM_BLOCKS * 8;
    const int wave0  = blockIdx.x * 8 + (threadIdx.x >> 5);

    for (int tile = wave0; tile < tiles; tile += gwaves) {
        const int r0 = tile * 16;
        int row = r0 + lan;
        if (row > n - 1) row = n - 1;     // clamp: keeps EXEC uniform
        const float* ap = X + (size_t)row * 64 + 2 * half;

        v8f accS[4] = {};
        v8f accD[4] = {};

#pragma unroll 2
        for (int t = 0; t < 16; ++t) {
            v2f a;
            a.x = ap[t * 4 + 0];
            a.y = ap[t * 4 + 1];
            const v2f* bS = &ldsS[(t * 64 + lan) * 2 + half];
            const v2f* bD = &ldsD[(t * 64 + lan) * 2 + half];
#pragma unroll
            for (int nt = 0; nt < 4; ++nt) {
                v2f bs = bS[nt * 32];   // (+16 cols) * 2 halves
                accS[nt] = __builtin_amdgcn_wmma_f32_16x16x4_f32(
                    false, a, false, bs, (short)0, accS[nt], false, false);
                v2f bd = bD[nt * 32];
                accD[nt] = __builtin_amdgcn_wmma_f32_16x16x4_f32(
                    false, a, false, bd, (short)0, accD[nt], false, false);
            }
        }

#pragma unroll
        for (int j = 0; j < 8; ++j) {
            const int rr = r0 + j + half * 8;
            if (rr < n) {
                const size_t base = (size_t)rr * 64 + lan;
#pragma unroll
                for (int nt = 0; nt < 4; ++nt) {
                    xs[base + nt * 16] = accS[nt][j];
                    xd[base + nt * 16] = accD[nt][j];
                }
            }
        }
    }
}

// al_s[n,h] = dot(xs[n, h*32 : h*32+32], a_s[h]); same for d-side.
__global__ void gat_alpha_coef(const float* __restrict__ xs,
                               const float* __restrict__ xd,
                               const float* __restrict__ a_s,
                               const float* __restrict__ a_d,
                               float* __restrict__ al_s,
                               float* __restrict__ al_d, int n) {
    int t = blockIdx.x * blockDim.x + threadIdx.x;
    if (t >= n * 2) return;
    const int node = t >> 1, h = t & 1;
    const float* pS = xs + (size_t)node * 64 + h * 32;
    const float* pD = xd + (size_t)node * 64 + h * 32;
    const float* aS = a_s + h * 32;
    const float* aD = a_d + h * 32;
    float ss = 0.0f, sd = 0.0f;
#pragma unroll
    for (int c = 0; c < 32; ++c) {
        ss += pS[c] * aS[c];
        sd += pD[c] * aD[c];
    }
    al_s[t] = ss;
    al_d[t] = sd;
}

// Monotonic float <-> uint key for atomicMax-based segment max.
__device__ __forceinline__ unsigned fkey(float f) {
    unsigned u = __float_as_uint(f);
    return (u & 0x80000000u) ? ~u : (u | 0x80000000u);
}
__device__ __forceinline__ float funkey(unsigned k) {
    unsigned u = (k & 0x80000000u) ? (k & 0x7fffffffu) : ~k;
    return __uint_as_float(u);
}
__device__ __forceinline__ float lrelu(float v) {
    return v > 0.0f ? v : 0.2f * v;
}

__global__ void gat_edge_max(const int* __restrict__ src,
                             const int* __restrict__ dst,
                             const float* __restrict__ al_s,
                             const float* __restrict__ al_d,
                             unsigned* __restrict__ mkey, int E) {
    int e = blockIdx.x * blockDim.x + threadIdx.x;
    if (e >= E) return;
    const int s = src[e], d = dst[e];
#pragma unroll
    for (int h = 0; h < 2; ++h) {
        float v = lrelu(al_s[s * 2 + h] + al_d[d * 2 + h]);
        atomicMax(&mkey[d * 2 + h], fkey(v));
    }
}

__global__ void gat_edge_expsum(const int* __restrict__ src,
                                const int* __restrict__ dst,
                                const float* __restrict__ al_s,
                                const float* __restrict__ al_d,
                                const unsigned* __restrict__ mkey,
                                float* __restrict__ ex,
                                float* __restrict__ den, int E) {
    int e = blockIdx.x * blockDim.x + threadIdx.x;
    if (e >= E) return;
    const int s = src[e], d = dst[e];
#pragma unroll
    for (int h = 0; h < 2; ++h) {
        float v = lrelu(al_s[s * 2 + h] + al_d[d * 2 + h]);
        float m = funkey(mkey[d * 2 + h]);
        float xv = __expf(v - m);
        ex[(size_t)e * 2 + h] = xv;
        atomicAdd(&den[d * 2 + h], xv);
    }
}

// 4 threads per edge; each covers 16 channels (one head-half) via float4.
__global__ void gat_aggregate(const int* __restrict__ src,
                              const int* __restrict__ dst,
                              const float* __restrict__ xs,
                              const float* __restrict__ ex,
                              const float* __restrict__ den,
                              float* __restrict__ acc, int E) {
    int t = blockIdx.x * blockDim.x + threadIdx.x;
    if (t >= E * 4) return;
    const int e = t >> 2, q = t & 3, h = q >> 1;
    const int s = src[e], d = dst[e];
    const float alpha = ex[(size_t)e * 2 + h] / den[d * 2 + h];
    const float4* xp = (const float4*)(xs + (size_t)s * 64 + q * 16);
    float* op = acc + (size_t)d * 64 + q * 16;
#pragma unroll
    for (int i = 0; i < 4; ++i) {
        float4 v = xp[i];
        atomicAdd(op + i * 4 + 0, v.x * alpha);
        atomicAdd(op + i * 4 + 1, v.y * alpha);
        atomicAdd(op + i * 4 + 2, v.z * alpha);
        atomicAdd(op + i * 4 + 3, v.w * alpha);
    }
}

__global__ void gat_bias_act(const float* __restrict__ acc,
                             const float* __restrict__ b,
                             float* __restrict__ out, int n, int doRelu) {
    int t = blockIdx.x * blockDim.x + threadIdx.x;
    if (t >= n * 64) return;
    float v = acc[t] + b[t & 63];
    if (doRelu) v = fmaxf(v, 0.0f);
    out[t] = v;
}

extern "C" void kernel_launch(void* const* d_in, const int* in_sizes, int n_in,
                              void* d_out, int out_size, void* d_ws, size_t ws_size,
                              hipStream_t stream) {
    const float* x  = (const float*)d_in[0];
    const int*   ei = (const int*)d_in[1];
    const int N = in_sizes[0] / 64;
    const int E = in_sizes[1] / 2;
    const int* src = ei;
    const int* dst = ei + E;

    // Workspace carve-up
    char* w = (char*)d_ws;
    const size_t NF = (size_t)N * 64 * sizeof(float);   // 25.6 MB
    const size_t EH = (size_t)E * 2 * sizeof(float);    // 12.8 MB
    const size_t NH = (size_t)N * 2 * sizeof(float);    // 0.8 MB
    float*    xs   = (float*)(w);
    float*    xd   = (float*)(w + NF);
    float*    bufA = (float*)(w + 2 * NF);
    float*    bufB = (float*)(w + 3 * NF);
    float*    ex   = (float*)(w + 4 * NF);
    float*    als  = (float*)(w + 4 * NF + EH);
    float*    ald  = (float*)(w + 4 * NF + EH + NH);
    unsigned* mkey = (unsigned*)(w + 4 * NF + EH + 2 * NH);
    float*    den  = (float*)(w + 4 * NF + EH + 3 * NH);

    const int coefBlocks = (N * 2 + 255) / 256;
    const int edgeBlocks = (E + 255) / 256;
    const int aggBlocks  = (E * 4 + 255) / 256;
    const int biasBlocks = (N * 64 + 255) / 256;

    for (int l = 0; l < 3; ++l) {
        const float* Xin = (l == 0) ? x : (l == 1) ? bufA : bufB;
        float* acc = (l == 0) ? bufA : (l == 1) ? bufB : (float*)d_out;
        const float* Ws  = (const float*)d_in[2 + l * 5];
        const float* Wd  = (const float*)d_in[3 + l * 5];
        const float* as_ = (const float*)d_in[4 + l * 5];
        const float* ad_ = (const float*)d_in[5 + l * 5];
        const float* b_  = (const float*)d_in[6 + l * 5];

        hipMemsetAsync(acc, 0, NF, stream);
        hipMemsetAsync(mkey, 0, NH, stream);
        hipMemsetAsync(den, 0, NH, stream);

        gat_gemm_wmma<<<GEMM_BLOCKS, 256, 0, stream>>>(Xin, Ws, Wd, xs, xd, N);
        gat_alpha_coef<<<coefBlocks, 256, 0, stream>>>(xs, xd, as_, ad_, als, ald, N);
        gat_edge_max<<<edgeBlocks, 256, 0, stream>>>(src, dst, als, ald, mkey, E);
        gat_edge_expsum<<<edgeBlocks, 256, 0, stream>>>(src, dst, als, ald, mkey, ex, den, E);
        gat_aggregate<<<aggBlocks, 256, 0, stream>>>(src, dst, xs, ex, den, acc, E);
        gat_bias_act<<<biasBlocks, 256, 0, stream>>>(acc, b_, acc, N, l < 2 ? 1 : 0);
    }
}